// DDAttention_88467736363078
// MI455X (gfx1250) — compile-verified
//
#include <hip/hip_runtime.h>

typedef __attribute__((ext_vector_type(2))) float v2f;
typedef __attribute__((ext_vector_type(8))) float v8f;

// Problem constants from the reference setup (fixed by setup_inputs()).
#define HEADS 8
#define NCH   32
#define NLVL  4
#define TILE  16        // queries per block tile (WMMA M)
#define JROWS 18        // 16 + 1 halo each side
#define KPAD  20        // JROWS padded to multiple of 4 for 16x16x4 WMMA
#define LDSW  33        // B row stride in floats (break 64-bank conflicts)
#define NPAIR (NLVL * JROWS)   // 72 (level, tap) pairs per tile

__global__ __launch_bounds__(64)
void msda_wmma_kernel(const float* __restrict__ v,
                      const float* __restrict__ sloc,
                      const float* __restrict__ aw,
                      const long long* __restrict__ shapes,
                      float* __restrict__ out,
                      int LQ, int LIN)
{
    __shared__ float  bmat_s[NLVL][KPAD][LDSW];  // sampled taps (B), padded K rows
    __shared__ float  amat_s[NLVL][TILE][KPAD];  // banded tap-weight matrix (A)
    __shared__ int4   coff_s[NPAIR];             // 4 corner offsets per (l, tap)
    __shared__ float4 cwgt_s[NPAIR];             // 4 corner blend weights

    const int tile = blockIdx.x;
    const int n    = blockIdx.y;               // 0..15  (= b*HEADS + head)
    const int b_i  = n / HEADS;
    const int head = n % HEADS;
    const int q0   = tile * TILE;
    const int t    = threadIdx.x;              // 0..63

    // Level geometry (tiny int64 shape table; 4 levels).
    int Hl[NLVL], Wl[NLVL], St[NLVL];
    {
        int s = 0;
        for (int l = 0; l < NLVL; ++l) {
            Hl[l] = (int)shapes[2 * l + 0];
            Wl[l] = (int)shapes[2 * l + 1];
            St[l] = s;
            s += Hl[l] * Wl[l];
        }
    }

    // Prefetch this tile's attention-weight rows (global_prefetch_b8).
    {
        int qp = q0 + (t & 15);
        if (qp >= LQ) qp = LQ - 1;
        __builtin_prefetch(&aw[(((size_t)b_i * LQ + qp) * HEADS + head) * 36], 0, 1);
    }

    // ---- Phase A: per-(level, tap) bilinear coords -> LDS (channel-independent) ----
    for (int p = t; p < NPAIR; p += 64) {
        const int l  = p / JROWS;
        const int j  = p - l * JROWS;
        const int qp = q0 - 1 + j;             // tap query index (zero-padded ends)

        int4   off = {0, 0, 0, 0};
        float4 wgt = {0.f, 0.f, 0.f, 0.f};
        if (qp >= 0 && qp < LQ) {
            const int sbase = (((b_i * LQ + qp) * HEADS + head) * NLVL + l) * 2;
            const float lx = sloc[sbase + 0];
            const float ly = sloc[sbase + 1];
            const int W_ = Wl[l], H_ = Hl[l];
            // align_corners=False: ix = lx*W - 0.5
            const float ix = lx * (float)W_ - 0.5f;
            const float iy = ly * (float)H_ - 0.5f;
            const float x0f = floorf(ix), y0f = floorf(iy);
            const int   x0 = (int)x0f,  y0 = (int)y0f;
            const int   x1 = x0 + 1,    y1 = y0 + 1;
            const float fx = ix - x0f,  fy = iy - y0f;
            const bool vx0 = (unsigned)x0 < (unsigned)W_;
            const bool vx1 = (unsigned)x1 < (unsigned)W_;
            const bool vy0 = (unsigned)y0 < (unsigned)H_;
            const bool vy1 = (unsigned)y1 < (unsigned)H_;
            // invalid corner -> weight 0, offset 0 (always a safe address)
            wgt.x = (vx0 && vy0) ? (1.0f - fx) * (1.0f - fy) : 0.0f;
            wgt.y = (vx1 && vy0) ? fx * (1.0f - fy)          : 0.0f;
            wgt.z = (vx0 && vy1) ? (1.0f - fx) * fy          : 0.0f;
            wgt.w = (vx1 && vy1) ? fx * fy                   : 0.0f;
            off.x = (vx0 && vy0) ? St[l] + y0 * W_ + x0 : 0;
            off.y = (vx1 && vy0) ? St[l] + y0 * W_ + x1 : 0;
            off.z = (vx0 && vy1) ? St[l] + y1 * W_ + x0 : 0;
            off.w = (vx1 && vy1) ? St[l] + y1 * W_ + x1 : 0;
        }
        coff_s[p] = off;
        cwgt_s[p] = wgt;
    }

    // ---- Phase B: banded A matrix -> LDS (one (level,row) per thread, full row) ----
    {
        const int l = t >> 4;                  // 0..3
        const int m = t & 15;                  // 0..15
        const int q = q0 + m;
        float w0 = 0.f, w1 = 0.f, w2 = 0.f;
        if (q < LQ) {
            const size_t abase = (((size_t)b_i * LQ + q) * HEADS + head) * 36;
            w0 = aw[abase +  4 + l];           // (3*0+1)*4 + l
            w1 = aw[abase + 16 + l];           // (3*1+1)*4 + l
            w2 = aw[abase + 28 + l];           // (3*2+1)*4 + l
        }
#pragma unroll
        for (int jj = 0; jj < KPAD; ++jj) {
            const int d = jj - m;
            amat_s[l][m][jj] = (d == 0) ? w0 : (d == 1) ? w1 : (d == 2) ? w2 : 0.0f;
        }
    }

    // ---- Zero the B K-padding rows (j = 18,19) ----
    for (int idx = t; idx < NLVL * 2 * NCH; idx += 64) {
        int l  = idx >> 6;
        int r  = (idx >> 5) & 1;
        int ch = idx & 31;
        bmat_s[l][JROWS + r][ch] = 0.0f;
    }

    __syncthreads();

    // ---- Phase C: branch-free gather, 4 levels x 18 taps x 32 channels ----
    for (int idx = t; idx < NPAIR * NCH; idx += 64) {
        const int p  = idx >> 5;               // (level, tap) pair 0..71
        const int ch = idx & 31;
        const int l  = p / JROWS;
        const int j  = p - l * JROWS;
        const int4   off = coff_s[p];
        const float4 wgt = cwgt_s[p];
        const float* vb = v + (size_t)(n * NCH + ch) * (size_t)LIN;
        float val = wgt.x * vb[off.x];
        val      += wgt.y * vb[off.y];
        val      += wgt.z * vb[off.z];
        val      += wgt.w * vb[off.w];
        bmat_s[l][j][ch] = val;
    }

    __syncthreads();

    // ---- Phase D: banded GEMM via V_WMMA_F32_16X16X4_F32 (pure LDS + WMMA) ----
    // Wave 0 -> channels 0..15, wave 1 -> channels 16..31.
    const int lane  = t & 31;
    const int m     = lane & 15;               // A row (M)
    const int khalf = lane >> 4;               // 0: K={0,1}, 1: K={2,3} of chunk
    const int ch    = (t >> 5) * 16 + (lane & 15);  // B/D column (N)

    v8f acc = {};
#pragma unroll
    for (int l = 0; l < NLVL; ++l) {
#pragma unroll
        for (int kc = 0; kc < KPAD / 4; ++kc) {
            const int j0 = kc * 4 + 2 * khalf;
            v2f a, b;
            a.x = amat_s[l][m][j0 + 0];
            a.y = amat_s[l][m][j0 + 1];
            b.x = bmat_s[l][j0 + 0][ch];
            b.y = bmat_s[l][j0 + 1][ch];
            acc = __builtin_amdgcn_wmma_f32_16x16x4_f32(
                    false, a, false, b, (short)0, acc, false, false);
        }
    }

    // ---- Phase E: scatter D (16x16 f32 layout: VGPR r -> M = r + 8*khalf) ----
#pragma unroll
    for (int r = 0; r < 8; ++r) {
        const int mq = r + 8 * khalf;
        const int qq = q0 + mq;
        if (qq < LQ) {
            out[(((size_t)b_i * LQ + qq) * HEADS + head) * NCH + ch] = acc[r];
        }
    }
}

extern "C" void kernel_launch(void* const* d_in, const int* in_sizes, int n_in,
                              void* d_out, int out_size, void* d_ws, size_t ws_size,
                              hipStream_t stream) {
    (void)n_in; (void)out_size; (void)d_ws; (void)ws_size;
    const float*     v      = (const float*)d_in[0];
    const float*     sloc   = (const float*)d_in[1];
    const float*     aw     = (const float*)d_in[2];
    const long long* shapes = (const long long*)d_in[3];
    float*           out    = (float*)d_out;

    const int B = 2;
    const int LQ  = in_sizes[1] / (B * HEADS * NLVL * 2);   // 5440
    const int LIN = in_sizes[0] / (B * HEADS * NCH);        // 5440

    const int ntiles = (LQ + TILE - 1) / TILE;              // 340
    dim3 grid(ntiles, B * HEADS);                           // 340 x 16 blocks
    msda_wmma_kernel<<<grid, 64, 0, stream>>>(v, sloc, aw, shapes, out, LQ, LIN);
}